// NodeModel_77884936945808
// MI455X (gfx1250) — compile-verified
//
#include <hip/hip_runtime.h>
#include <hip/hip_bf16.h>

typedef __attribute__((ext_vector_type(2))) float v2f;
typedef __attribute__((ext_vector_type(4))) float v4f;
typedef __attribute__((ext_vector_type(8))) float v8f;

#define C2   240     // conv3 out channels (15 tiles of 16)
#define EDG  8192    // K dimension
#define LEN  1024    // N dimension (64 tiles of 16)

// ---------------------------------------------------------------------------
// K1: edge_partial[s, 240, 1024] = conv3_w[mtile] @ edge_attr[ksplit, ntile]
// One wave32 per (m, n, s) tile. WMMA f32 16x16x4, fp32-exact.
// ---------------------------------------------------------------------------
__global__ __launch_bounds__(256) void gemm_splitk_wmma(
    const float* __restrict__ w3,       // [240, 8192] row-major
    const float* __restrict__ ea,       // [8192, 1024] row-major
    float* __restrict__ partial,        // [S, 240, 1024]
    int S)
{
    const int wid     = blockIdx.x * 8 + (threadIdx.x >> 5);   // 8 waves/block
    const int lane    = threadIdx.x & 31;
    const int s       = wid % S;
    const int n       = (wid / S) & 63;          // 64 N tiles
    const int m       = wid / (S * 64);          // 15 M tiles
    const int half    = lane >> 4;               // 0: K={0,1}, 1: K={2,3}
    const int lane16  = lane & 15;

    const int kPerSplit = EDG / S;
    const int k0        = s * kPerSplit;

    // A (16x4, MxK): lane holds row M=lane16, K pair selected by half.
    const float* aPtr = w3 + (size_t)(m * 16 + lane16) * EDG + k0 + half * 2;
    // B (4x16, KxN): lane holds col N=lane16, K rows {2*half, 2*half+1}.
    const float* bPtr = ea + (size_t)(k0 + half * 2) * LEN + n * 16 + lane16;

    v8f c = {0.f, 0.f, 0.f, 0.f, 0.f, 0.f, 0.f, 0.f};

#pragma unroll 4
    for (int k = 0; k < kPerSplit; k += 4) {
        v2f a, b;
        a.x = aPtr[0];
        a.y = aPtr[1];
        b.x = bPtr[0];
        b.y = bPtr[LEN];
        aPtr += 4;
        bPtr += 4 * LEN;
        // 8 args: (neg_a, A, neg_b, B, c_mod, C, reuse_a, reuse_b)
        c = __builtin_amdgcn_wmma_f32_16x16x4_f32(
                false, a, false, b, (short)0, c, false, false);
    }

    // C/D layout: VGPR v -> row M = v + 8*half, col N = lane16.
    float* out = partial + ((size_t)s * C2 + m * 16 + 8 * half) * LEN
                         + n * 16 + lane16;
#pragma unroll
    for (int v = 0; v < 8; ++v)
        out[(size_t)v * LEN] = c[v];
}

// ---------------------------------------------------------------------------
// K2: per-column split-K reduce + bias + relu, fused conv4 dot + sigmoid +
//     scaled column sum.  col[n] = (1/LEN) * (1+sigmoid(t_n)) * sum_m edge
// ---------------------------------------------------------------------------
__global__ __launch_bounds__(256) void reduce_mask_colsum(
    const float* __restrict__ partial,  // [S, 240, 1024]
    const float* __restrict__ b3,       // [240]
    const float* __restrict__ w4,       // [240]
    const float* __restrict__ b4,       // [1]
    float* __restrict__ col,            // [1024]
    int S)
{
    const int n = blockIdx.x * blockDim.x + threadIdx.x;
    if (n >= LEN) return;

    float acc4 = b4[0];
    float accs = 0.f;
    for (int m = 0; m < C2; ++m) {
        float v = b3[m];
        for (int s = 0; s < S; ++s)
            v += partial[((size_t)s * C2 + m) * LEN + n];
        v = fmaxf(v, 0.f);
        acc4 = fmaf(w4[m], v, acc4);
        accs += v;
    }
    const float mask = 1.f / (1.f + __expf(-acc4));
    col[n] = accs * (1.f + mask) * (1.f / (float)LEN);
}

// ---------------------------------------------------------------------------
// K3: broadcast col[1024] to all rows. 40 MB write-once -> non-temporal
//     stores so edge_attr/conv3_w stay resident in L2 for the next replay.
// ---------------------------------------------------------------------------
__global__ __launch_bounds__(256) void broadcast_rows(
    const v4f* __restrict__ col4,       // [256]
    v4f* __restrict__ out4)             // [rows, 256]
{
    const v4f v = col4[threadIdx.x];               // L1/L2 hit after warmup
    __builtin_nontemporal_store(v, &out4[(size_t)blockIdx.x * 256 + threadIdx.x]);
}

// ---------------------------------------------------------------------------
extern "C" void kernel_launch(void* const* d_in, const int* in_sizes, int n_in,
                              void* d_out, int out_size, void* d_ws, size_t ws_size,
                              hipStream_t stream) {
    // setup_inputs order: x, edge_index, edge_attr, conv3_w, conv3_b, conv4_w, conv4_b
    const float* ea = (const float*)d_in[2];
    const float* w3 = (const float*)d_in[3];
    const float* b3 = (const float*)d_in[4];
    const float* w4 = (const float*)d_in[5];
    const float* b4 = (const float*)d_in[6];
    float* out = (float*)d_out;

    const int rows = out_size / LEN;               // 10000

    // Pick the largest split-K that fits in workspace.
    int S = 1;
    for (int cand = 8; cand >= 1; cand >>= 1) {
        size_t need = ((size_t)cand * C2 * LEN + LEN) * sizeof(float);
        if (need <= ws_size) { S = cand; break; }
    }

    float* partial = (float*)d_ws;                         // [S,240,1024]
    float* col     = partial + (size_t)S * C2 * LEN;       // [1024]

    // K1: 15 * 64 * S waves, 8 waves per 256-thread block.
    const int waves  = 15 * 64 * S;
    const int blocks = waves / 8;
    gemm_splitk_wmma<<<blocks, 256, 0, stream>>>(w3, ea, partial, S);

    // K2: one thread per column.
    reduce_mask_colsum<<<(LEN + 255) / 256, 256, 0, stream>>>(
        partial, b3, w4, b4, col, S);

    // K3: one block per output row (256 float4 = 1024 floats).
    broadcast_rows<<<rows, 256, 0, stream>>>((const v4f*)col, (v4f*)out);
}